// Model_64321430225212
// MI455X (gfx1250) — compile-verified
//
#include <hip/hip_runtime.h>
#include <hip/hip_bf16.h>

typedef __attribute__((ext_vector_type(2))) float v2f;
typedef __attribute__((ext_vector_type(8))) float v8f;

// Only meaningful in the device pass; host pass (x86) never has amdgcn builtins.
#if defined(__HIP_DEVICE_COMPILE__)
#if !__has_builtin(__builtin_amdgcn_wmma_f32_16x16x4_f32)
#error "builtin __builtin_amdgcn_wmma_f32_16x16x4_f32 not available for this offload target"
#endif
#endif

#define N_PTS 2048
#define DIM   128
#define TILES_PER_DIM      (N_PTS / 16)                       // 128
#define TILES_PER_PAIRMAT  (TILES_PER_DIM * TILES_PER_DIM)    // 16384
#define WAVES_PER_TB       8
#define TB_PER_PAIRMAT     (TILES_PER_PAIRMAT / WAVES_PER_TB) // 2048
#define NUM_TB             (3 * TB_PER_PAIRMAT)               // 6144

// 1/(2*sigma^2) for sigma = 0.1, 1.0, 10.0
#define INV0 50.0f
#define INV1 0.5f
#define INV2 0.005f

// ---------------------------------------------------------------------------
// Stage 1: row squared-norms for source (i in [0,2048)) and target ([2048,4096))
// ---------------------------------------------------------------------------
__global__ void mmd_norms_kernel(const float* __restrict__ src,
                                 const float* __restrict__ tgt,
                                 float* __restrict__ norms) {
  int i = blockIdx.x * blockDim.x + threadIdx.x;        // 0..4095
  const float* base = (i < N_PTS) ? src : tgt;
  int row = i & (N_PTS - 1);
  const float4* p = (const float4*)(base + row * DIM);
  float s = 0.0f;
#pragma unroll
  for (int j = 0; j < DIM / 4; ++j) {
    float4 v = p[j];
    s += v.x * v.x + v.y * v.y + v.z * v.z + v.w * v.w;
  }
  norms[i] = s;
}

// ---------------------------------------------------------------------------
// Stage 2: one wave computes one 16x16 tile of one pairwise kernel matrix via
// chained V_WMMA_F32_16X16X4_F32, fuses dist^2 + three exps, reduces.
// Pair-matrix id pb: 0 = K_ss, 1 = K_tt, 2 = K_st.
// ---------------------------------------------------------------------------
__global__ void __launch_bounds__(256)
mmd_pair_kernel(const float* __restrict__ S, const float* __restrict__ T,
                const float* __restrict__ norms, float* __restrict__ partials) {
  const int lane = threadIdx.x & 31;
  const int wave = threadIdx.x >> 5;

  const int gtile = blockIdx.x * WAVES_PER_TB + wave;   // 0..49151
  const int pb = gtile / TILES_PER_PAIRMAT;             // all waves in a TB share pb
  const int t  = gtile % TILES_PER_PAIRMAT;
  const int tm = t >> 7;          // tile row (128 tiles per dim)
  const int tn = t & 127;         // tile col

  const float* X  = (pb == 1) ? T : S;                  // rows of the tile
  const float* Y  = (pb == 0) ? S : T;                  // cols of the tile
  const float* nx = norms + ((pb == 1) ? N_PTS : 0);
  const float* ny = norms + ((pb == 0) ? 0 : N_PTS);

  // A layout (f32 16x4): lane m = lane&15; VGPR pair holds K = 2*(lane>>4), +1
  // B layout mirrors with column n = lane&15 of Y^T, i.e. row n of Y.
  const int mn  = lane & 15;
  const int kh  = (lane >> 4) << 1;                     // 0 or 2
  const float* arow = X + (tm * 16 + mn) * DIM + kh;
  const float* brow = Y + (tn * 16 + mn) * DIM + kh;

  v8f acc = {};
#pragma unroll
  for (int k0 = 0; k0 < DIM; k0 += 4) {
    v2f a = *(const v2f*)(arow + k0);                   // 8B aligned (kh even, k0%4==0)
    v2f b = *(const v2f*)(brow + k0);
    acc = __builtin_amdgcn_wmma_f32_16x16x4_f32(
        /*neg_a=*/false, a, /*neg_b=*/false, b,
        /*c_mod=*/(short)0, acc, /*reuse_a=*/false, /*reuse_b=*/false);
  }

  // C/D layout: acc[v] = dot tile element (M = v + 8*(lane>>4), N = lane&15)
  const float yyn  = ny[tn * 16 + mn];
  const int  mbase = tm * 16 + ((lane >> 4) << 3);
  float s0 = 0.0f, s1 = 0.0f, s2 = 0.0f;
#pragma unroll
  for (int v = 0; v < 8; ++v) {
    float d2 = nx[mbase + v] + yyn - 2.0f * acc[v];
    d2 = fmaxf(d2, 0.0f);
    s0 += __expf(-INV0 * d2);
    s1 += __expf(-INV1 * d2);
    s2 += __expf(-INV2 * d2);
  }

  // wave32 reduction (deterministic fixed order)
#pragma unroll
  for (int off = 16; off > 0; off >>= 1) {
    s0 += __shfl_down(s0, off, 32);
    s1 += __shfl_down(s1, off, 32);
    s2 += __shfl_down(s2, off, 32);
  }

  __shared__ float lred[WAVES_PER_TB][3];
  if (lane == 0) { lred[wave][0] = s0; lred[wave][1] = s1; lred[wave][2] = s2; }
  __syncthreads();

  if (threadIdx.x < 3) {
    float acc3 = 0.0f;
#pragma unroll
    for (int w = 0; w < WAVES_PER_TB; ++w) acc3 += lred[w][threadIdx.x];
    partials[blockIdx.x * 3 + threadIdx.x] = acc3;
  }
}

// ---------------------------------------------------------------------------
// Stage 3: deterministic final reduction + MMD formula.
// partials layout: block b (pb = b / 2048) wrote partials[b*3 + sigma].
// ---------------------------------------------------------------------------
__global__ void mmd_finalize_kernel(const float* __restrict__ partials,
                                    float* __restrict__ out) {
  __shared__ float red[256];
  float sums[9];

  for (int idx = 0; idx < 9; ++idx) {
    const int pb = idx / 3, sg = idx % 3;
    float acc = 0.0f;
    for (int b = threadIdx.x; b < TB_PER_PAIRMAT; b += 256)
      acc += partials[(pb * TB_PER_PAIRMAT + b) * 3 + sg];
    red[threadIdx.x] = acc;
    __syncthreads();
    for (int off = 128; off > 0; off >>= 1) {
      if ((int)threadIdx.x < off) red[threadIdx.x] += red[threadIdx.x + off];
      __syncthreads();
    }
    sums[idx] = red[0];
    __syncthreads();
  }

  if (threadIdx.x == 0) {
    const float n = (float)N_PTS;
    const float denom = n * (n - 1.0f);
    float total = 0.0f;
    for (int sg = 0; sg < 3; ++sg) {
      // trace(K_ss) == trace(K_tt) == n exactly (exp(0) on the diagonal)
      float mmd = (sums[0 * 3 + sg] - n) / denom
                + (sums[1 * 3 + sg] - n) / denom
                - 2.0f * sums[2 * 3 + sg] / (n * n);
      total += mmd;
    }
    out[0] = total / 3.0f;
  }
}

// ---------------------------------------------------------------------------
extern "C" void kernel_launch(void* const* d_in, const int* in_sizes, int n_in,
                              void* d_out, int out_size, void* d_ws, size_t ws_size,
                              hipStream_t stream) {
  (void)in_sizes; (void)n_in; (void)out_size; (void)ws_size;

  const float* src = (const float*)d_in[0];   // (2048, 128) fp32
  const float* tgt = (const float*)d_in[1];   // (2048, 128) fp32

  float* ws       = (float*)d_ws;
  float* norms    = ws;                       // 4096 floats
  float* partials = ws + 2 * N_PTS;           // 6144 * 3 floats

  mmd_norms_kernel<<<(2 * N_PTS) / 256, 256, 0, stream>>>(src, tgt, norms);
  mmd_pair_kernel<<<NUM_TB, 256, 0, stream>>>(src, tgt, norms, partials);
  mmd_finalize_kernel<<<1, 256, 0, stream>>>(partials, (float*)d_out);
}